// maxpool_neuron_18622978195624
// MI455X (gfx1250) — compile-verified
//
#include <hip/hip_runtime.h>

// ---------------------------------------------------------------------------
// Problem constants (from the reference): x is [(T+1)*N, C, H, W] f32,
// out is [(T+1)*N, C, H/2, W/2] f32, T=8, N=8, C=64, H=W=112.
// Memory-bound streaming recurrence: ~289 MB total traffic -> ~12.4 us floor
// at 23.3 TB/s. No GEMM structure, so no WMMA; the CDNA5 win is async
// global->LDS double buffering of the serial temporal loop.
// ---------------------------------------------------------------------------

namespace {
constexpr int T_STEPS = 9;     // T+1 frames
constexpr int NB      = 8;     // batch N
constexpr int C       = 64;
constexpr int H       = 112;
constexpr int W       = 112;
constexpr int HP      = H / 2;
constexpr int WP      = W / 2;

constexpr int TILE_ROWS   = 8;                 // input rows per workgroup tile
constexpr int TILE_WROWS  = TILE_ROWS / 2;     // 4 pooled rows
constexpr int THREADS     = TILE_WROWS * WP;   // 224 threads = 7 wave32
constexpr int TILE_FLOATS = TILE_ROWS * W;     // 896 floats = 3584 B (contiguous)

constexpr long long FRAME_STRIDE     = (long long)NB * C * H * W;    // floats
constexpr long long OUT_FRAME_STRIDE = (long long)NB * C * HP * WP;  // floats
}

// Async DMA of 16B/lane from global into LDS (CDNA5 GLOBAL_LOAD_ASYNC_TO_LDS_B128,
// tracked by ASYNCcnt). lds_byte_addr = low 32 bits of the flat shared pointer
// (== workgroup-relative LDS byte offset per the gfx1250 aperture rules).
__device__ __forceinline__ void async_load_b128(unsigned lds_byte_addr,
                                                const float* gaddr) {
  asm volatile("global_load_async_to_lds_b128 %0, %1, off"
               :
               : "v"(lds_byte_addr), "v"(gaddr)
               : "memory");
}

extern "C" __global__ __launch_bounds__(THREADS)
void snn_diff_maxpool_kernel(const float* __restrict__ x,
                             float* __restrict__ out) {
  __shared__ __align__(16) float lds[2][TILE_FLOATS];

  const int tid  = threadIdx.x;
  const int tile = blockIdx.x;   // 0..13 row-tiles of 8 input rows
  const int c    = blockIdx.y;   // 0..63
  const int n    = blockIdx.z;   // 0..7

  const long long planeBase =
      ((long long)n * C + c) * (H * W) + (long long)tile * TILE_ROWS * W;

  // Each lane copies one contiguous 16B chunk of the 3584B tile per frame.
  const float*   gsrc   = x + planeBase + 4 * tid;
  const unsigned lbase0 = (unsigned)(size_t)(&lds[0][4 * tid]);
  const unsigned lbase1 = (unsigned)(size_t)(&lds[1][4 * tid]);

  // Prime the double buffer: frames 0 and 1 in flight.
  async_load_b128(lbase0, gsrc);
  async_load_b128(lbase1, gsrc + FRAME_STRIDE);

  const int wr = tid / WP;           // pooled row within tile: 0..3
  const int wc = tid - wr * WP;      // pooled col: 0..55
  const int r0 = (2 * wr) * W + 2 * wc;
  const int r1 = r0 + W;

  const long long outPlane =
      ((long long)n * C + c) * (HP * WP) + (tile * TILE_WROWS + wr) * WP + wc;

  float b00 = 0.f, b01 = 0.f, b10 = 0.f, b11 = 0.f;   // bias = frame 0
  float e00 = 0.f, e01 = 0.f, e10 = 0.f, e11 = 0.f;   // exp_in
  float v00 = 0.f, v01 = 0.f, v10 = 0.f, v11 = 0.f;   // membrane
  float eout = 0.f;                                   // exp_out

  for (int t = 0; t < T_STEPS; ++t) {
    // Async loads complete in order: cnt<=1 guarantees frame t has landed
    // while frame t+1 may still be in flight; last frame needs cnt==0.
    if (t + 1 < T_STEPS) {
      asm volatile("s_wait_asynccnt 0x1" ::: "memory");
    } else {
      asm volatile("s_wait_asynccnt 0x0" ::: "memory");
    }
    __syncthreads();   // all 7 waves' chunks of frame t are visible in LDS

    const float* buf = lds[t & 1];
    const float a00 = buf[r0];
    const float a01 = buf[r0 + 1];
    const float a10 = buf[r1];
    const float a11 = buf[r1 + 1];

    float o;
    if (t == 0) {
      b00 = a00; b01 = a01; b10 = a10; b11 = a11;   // bias
      e00 = a00; e01 = a01; e10 = a10; e11 = a11;   // exp_in0
      o = 0.f;                                      // exp_out0 frame
    } else {
      const float tf = (float)t;
      const float d00 = a00 - b00, d01 = a01 - b01;
      const float d10 = a10 - b10, d11 = a11 - b11;
      v00 += d00 + e00; v01 += d01 + e01;
      v10 += d10 + e10; v11 += d11 + e11;
      e00 += d00 / tf;  e01 += d01 / tf;
      e10 += d10 / tf;  e11 += d11 / tf;
      const float m = fmaxf(fmaxf(v00, v01), fmaxf(v10, v11));  // maxpool 2x2
      v00 -= m; v01 -= m; v10 -= m; v11 -= m;                   // v -= unpool(o)
      o = m - eout;
      eout += o / tf;
    }

    // Write-once streaming output: bypass cache retention.
    __builtin_nontemporal_store(
        o, out + (long long)t * OUT_FRAME_STRIDE + outPlane);

    __syncthreads();   // every wave finished reading buf[t&1]
    if (t + 2 < T_STEPS) {
      const unsigned ldst = (t & 1) ? lbase1 : lbase0;
      async_load_b128(ldst, gsrc + (long long)(t + 2) * FRAME_STRIDE);
    }
  }
}

extern "C" void kernel_launch(void* const* d_in, const int* in_sizes, int n_in,
                              void* d_out, int out_size, void* d_ws,
                              size_t ws_size, hipStream_t stream) {
  (void)in_sizes; (void)n_in; (void)out_size; (void)d_ws; (void)ws_size;
  const float* x  = (const float*)d_in[0];
  float* outp     = (float*)d_out;
  dim3 grid(H / TILE_ROWS, C, NB);   // 14 x 64 x 8 = 7168 workgroups
  snn_diff_maxpool_kernel<<<grid, dim3(THREADS), 0, stream>>>(x, outp);
}